// NeighborPoolingLayer_90357521973574
// MI455X (gfx1250) — compile-verified
//
#include <hip/hip_runtime.h>

typedef __attribute__((ext_vector_type(2))) float v2f;
typedef __attribute__((ext_vector_type(8))) float v8f;

#define CCH 64              // channels
#define ROWS_PER_WAVE 16    // one 16-row WMMA tile per wave
#define WAVES_PER_BLOCK 8   // 256 threads

__global__ __launch_bounds__(WAVES_PER_BLOCK * 32)
void neighbor_mean_wmma(const float* __restrict__ feat,
                        const int*   __restrict__ nbr,
                        const int*   __restrict__ rs,
                        float*       __restrict__ out,
                        int M, int E)
{
    const int lane = threadIdx.x & 31;
    const int wave = threadIdx.x >> 5;
    const int m0   = (blockIdx.x * WAVES_PER_BLOCK + wave) * ROWS_PER_WAVE;
    if (m0 >= M) return;                       // wave-uniform exit, EXEC stays full

    const int r    = lane & 15;                // local output row 0..15
    const int half = lane >> 4;                // half-wave selects K pair
    const int k_lo = half * 2;                 // K offset handled by this half
    const int n    = lane & 15;                // channel-in-tile

    // Per-row CSR span and inverse count (lanes 16-31 mirror lanes 0-15).
    const int row     = min(m0 + r, M - 1);
    const int start_r = rs[row];
    const int end_r   = rs[row + 1];
    const float inv   = 1.0f / fmaxf((float)(end_r - start_r), 1.0f);

    // Wave-uniform edge span covering all 16 rows of this tile.
    int spanStart = __builtin_amdgcn_readfirstlane(rs[m0]);
    int spanEnd   = __builtin_amdgcn_readfirstlane(rs[min(m0 + ROWS_PER_WAVE, M)]);

    v8f acc0 = {}, acc1 = {}, acc2 = {}, acc3 = {};

    // Software-pipelined neighbor indices (one 4-edge chunk ahead).
    int e = spanStart;
    int i0 = 0, i1 = 0;
    if (e < spanEnd) {
        const int e0 = e + k_lo;
        i0 = nbr[min(e0,     E - 1)];
        i1 = nbr[min(e0 + 1, E - 1)];
    }

    for (; e < spanEnd; e += 4) {              // uniform loop bounds
        const int e0  = e + k_lo;
        const int e1  = e0 + 1;
        const int ci0 = i0, ci1 = i1;

        // Preload next chunk's indices and prefetch its feature rows
        // (covers all 4 rows x 256B across the wave's lanes).
        const int en = e + 4;
        if (en < spanEnd) {                    // uniform branch
            const int ne0 = en + k_lo;
            i0 = nbr[min(ne0,     E - 1)];
            i1 = nbr[min(ne0 + 1, E - 1)];
            const char* p0 = (const char*)(feat + (size_t)i0 * CCH) + n * 16;
            const char* p1 = (const char*)(feat + (size_t)i1 * CCH) + n * 16;
            __builtin_prefetch(p0, 0, 3);      // -> global_prefetch_b8
            __builtin_prefetch(p1, 0, 3);
        }

        // A (16x4 f32): VGPR0 = weight of edge K=k_lo, VGPR1 = K=k_lo+1.
        const float w0 = (e0 >= start_r && e0 < end_r) ? inv : 0.0f;
        const float w1 = (e1 >= start_r && e1 < end_r) ? inv : 0.0f;
        const v2f A = { w0, w1 };

        const float* f0 = feat + (size_t)ci0 * CCH;
        const float* f1 = feat + (size_t)ci1 * CCH;

        // B (4x16 f32) for the 4 channel tiles; lanes 0-15 coalesce to 64B rows.
        const v2f B0 = { f0[n],      f1[n]      };
        const v2f B1 = { f0[16 + n], f1[16 + n] };
        const v2f B2 = { f0[32 + n], f1[32 + n] };
        const v2f B3 = { f0[48 + n], f1[48 + n] };

        acc0 = __builtin_amdgcn_wmma_f32_16x16x4_f32(false, A, false, B0, (short)0, acc0, false, false);
        acc1 = __builtin_amdgcn_wmma_f32_16x16x4_f32(false, A, false, B1, (short)0, acc1, false, false);
        acc2 = __builtin_amdgcn_wmma_f32_16x16x4_f32(false, A, false, B2, (short)0, acc2, false, false);
        acc3 = __builtin_amdgcn_wmma_f32_16x16x4_f32(false, A, false, B3, (short)0, acc3, false, false);
    }

    // C/D layout: VGPR j -> row m0 + j + 8*half, column n (per 16-wide tile).
    const int orow = m0 + 8 * half;
    float* obase = out + (size_t)orow * CCH + n;
#pragma unroll
    for (int j = 0; j < 8; ++j) {
        if (orow + j < M) {
            float* po = obase + (size_t)j * CCH;
            po[0]  = acc0[j];
            po[16] = acc1[j];
            po[32] = acc2[j];
            po[48] = acc3[j];
        }
    }
}

extern "C" void kernel_launch(void* const* d_in, const int* in_sizes, int n_in,
                              void* d_out, int out_size, void* d_ws, size_t ws_size,
                              hipStream_t stream) {
    const float* feat = (const float*)d_in[0];   // [N, 64] f32
    const int*   nbr  = (const int*)d_in[1];     // [E] i32
    const int*   rs   = (const int*)d_in[2];     // [M+1] i32
    float*       out  = (float*)d_out;           // [M, 64] f32

    const int E = in_sizes[1];
    const int M = in_sizes[2] - 1;

    const int rowsPerBlock = ROWS_PER_WAVE * WAVES_PER_BLOCK;   // 128
    const int grid = (M + rowsPerBlock - 1) / rowsPerBlock;     // 1024 for M=131072

    hipLaunchKernelGGL(neighbor_mean_wmma, dim3(grid), dim3(WAVES_PER_BLOCK * 32),
                       0, stream, feat, nbr, rs, out, M, E);
}